// SAGEConv_bigraph_1872605741717
// MI455X (gfx1250) — compile-verified
//
#include <hip/hip_runtime.h>

#define D 128             // D_IN == D_OUT == 128
#define AROWS 132         // padded LDS row stride (floats): 16B-aligned rows,
                          // bank stride 4 -> conflict-free ds_load_b64

typedef __attribute__((ext_vector_type(2))) float v2f;
typedef __attribute__((ext_vector_type(8))) float v8f;

// ---------------------------------------------------------------------------
// Inline CDNA5 helpers
// ---------------------------------------------------------------------------
// Non-returning hardware f32 atomic add, device scope (resolves in L2).
__device__ __forceinline__ void atomic_add_f32_dev(float* p, float v) {
    asm volatile("global_atomic_add_f32 %0, %1, off scope:SCOPE_DEV"
                 :: "v"(p), "v"(v) : "memory");
}

// Async global->LDS copy of 16 bytes (tracked by ASYNCcnt).
__device__ __forceinline__ void async_copy_b128(unsigned lds_off, const void* gptr) {
    asm volatile("global_load_async_to_lds_b128 %0, %1, off"
                 :: "v"(lds_off), "v"(gptr) : "memory");
}

__device__ __forceinline__ void wait_asynccnt0() {
    asm volatile("s_wait_asynccnt 0x0" ::: "memory");
}

// ---------------------------------------------------------------------------
// Phase 0: zero workspace (summed[n_dst*128] + deg[n_dst]) every call.
// ---------------------------------------------------------------------------
__global__ void zero_ws_kernel(float4* __restrict__ ws, int n4) {
    int i = blockIdx.x * blockDim.x + threadIdx.x;
    int stride = gridDim.x * blockDim.x;
    const float4 z = {0.f, 0.f, 0.f, 0.f};
    for (; i < n4; i += stride) ws[i] = z;
}

// ---------------------------------------------------------------------------
// Phase 1: edge scatter. One wave (32 lanes) per edge; each lane moves a
// float4 (32 lanes * 4 = 128 floats = one feature row). feat_src (51.2 MB)
// and summed (10.2 MB) are L2-resident in the 192MB L2, so the non-returning
// global_atomic_add_f32 (scope DEV) resolves in L2 at line rate.
// ---------------------------------------------------------------------------
__global__ __launch_bounds__(256) void edge_scatter_kernel(
        const float* __restrict__ feat_src,
        const int*   __restrict__ src_idx,
        const int*   __restrict__ dst_idx,
        float*       __restrict__ summed,
        float*       __restrict__ deg,
        int n_edges) {
    int edge = (blockIdx.x * blockDim.x + threadIdx.x) >> 5;
    int lane = threadIdx.x & 31;
    if (edge >= n_edges) return;

    int s = src_idx[edge];
    int d = dst_idx[edge];

    const float4 v = ((const float4*)(feat_src + (size_t)s * D))[lane];
    float* dstp = summed + (size_t)d * D + lane * 4;
    atomic_add_f32_dev(dstp + 0, v.x);
    atomic_add_f32_dev(dstp + 1, v.y);
    atomic_add_f32_dev(dstp + 2, v.z);
    atomic_add_f32_dev(dstp + 3, v.w);
    if (lane == 0) atomic_add_f32_dev(deg + d, 1.0f);
}

// ---------------------------------------------------------------------------
// Phase 2: out = feat_dst @ W_self^T + (summed/deg) @ W_neigh^T + biases
// via V_WMMA_F32_16X16X4_F32. One wave per 16x16 output tile:
//   blockIdx.x    -> row tile (16 rows), 20000/16 = 1250 blocks
//   wave-in-block -> col tile (16 cols), 128/16  = 8 waves/block
// The 16x128 A-panels (feat_dst and summed) are shared by all 8 waves, so
// they are staged once per block into LDS with GLOBAL_LOAD_ASYNC_TO_LDS_B128
// (ASYNCcnt + s_wait_asynccnt), then read conflict-free via ds_load_b64.
// f32 WMMA operand layouts (ISA 7.12.2):
//   A 16x4 : lane l -> row (l&15), K = 2*(l>>4)+{0,1}   (contiguous float2)
//   B 4x16 : lane l -> col (l&15), K = 2*(l>>4)+{0,1}   (contiguous float2
//            of row-major W[out,in] since B = W^T)
//   C/D    : vgpr j -> row rm + j + 8*(l>>4), col cn + (l&15)
// ---------------------------------------------------------------------------
__global__ __launch_bounds__(256) void sage_wmma_kernel(
        const float* __restrict__ feat_dst,
        const float* __restrict__ summed,
        const float* __restrict__ deg,
        const float* __restrict__ W_self,
        const float* __restrict__ b_self,
        const float* __restrict__ W_neigh,
        const float* __restrict__ b_neigh,
        float*       __restrict__ out) {
    __shared__ float lds_xd[16 * AROWS];   // feat_dst panel (padded rows)
    __shared__ float lds_xn[16 * AROWS];   // summed   panel (padded rows)

    const int tid  = threadIdx.x;        // 0..255
    const int lane = tid & 31;
    const int wv   = tid >> 5;           // 0..7 -> column tile
    const int lo   = lane & 15;
    const int hi   = lane >> 4;          // 0 or 1
    const int rm   = blockIdx.x * 16;    // output row base
    const int cn   = wv * 16;            // output col base

    // ---- async stage of both 16x128 A-panels into LDS ----
    // 2 panels * 16 rows * 8 float4 = 256 B128 transfers -> one per thread.
    {
        const int chunk = tid & 127;     // 0..127
        const int row   = chunk >> 3;    // 0..15
        const int c4    = chunk & 7;     // float4 index within row
        const float* gsrc = (tid < 128 ? feat_dst : summed)
                            + (size_t)(rm + row) * D + c4 * 4;
        const float* ldst = (tid < 128 ? lds_xd : lds_xn)
                            + row * AROWS + c4 * 4;
        async_copy_b128((unsigned)(uintptr_t)ldst, gsrc);
    }
    wait_asynccnt0();
    __syncthreads();

    const int bcol = cn + lo;            // B-operand column handled by this lane
    const float* arow_d = lds_xd + lo * AROWS;
    const float* arow_n = lds_xn + lo * AROWS;

    v8f acc = {};

    // ---- self term: A = feat_dst (LDS), B = W_self^T ----
#pragma unroll 8
    for (int kb = 0; kb < D; kb += 4) {
        const int k = kb + 2 * hi;
        v2f a = *(const v2f*)(arow_d + k);
        v2f b = *(const v2f*)(W_self + (size_t)bcol * D + k);
        acc = __builtin_amdgcn_wmma_f32_16x16x4_f32(
                  false, a, false, b, (short)0, acc, false, false);
    }

    // ---- neighbor term: A = summed * (1/max(deg,1)) (LDS), B = W_neigh^T ----
    const float scale = 1.0f / fmaxf(deg[rm + lo], 1.0f);
#pragma unroll 8
    for (int kb = 0; kb < D; kb += 4) {
        const int k = kb + 2 * hi;
        v2f a = *(const v2f*)(arow_n + k);
        a.x *= scale;
        a.y *= scale;
        v2f b = *(const v2f*)(W_neigh + (size_t)bcol * D + k);
        acc = __builtin_amdgcn_wmma_f32_16x16x4_f32(
                  false, a, false, b, (short)0, acc, false, false);
    }

    // ---- store with fused bias ----
    const float bias = b_self[bcol] + b_neigh[bcol];
#pragma unroll
    for (int j = 0; j < 8; ++j) {
        const int m = rm + j + 8 * hi;
        out[(size_t)m * D + bcol] = acc[j] + bias;
    }
}

// ---------------------------------------------------------------------------
extern "C" void kernel_launch(void* const* d_in, const int* in_sizes, int n_in,
                              void* d_out, int out_size, void* d_ws, size_t ws_size,
                              hipStream_t stream) {
    const float* feat_src = (const float*)d_in[0];
    const float* feat_dst = (const float*)d_in[1];
    const int*   src_idx  = (const int*)  d_in[2];
    const int*   dst_idx  = (const int*)  d_in[3];
    const float* W_self   = (const float*)d_in[4];
    const float* b_self   = (const float*)d_in[5];
    const float* W_neigh  = (const float*)d_in[6];
    const float* b_neigh  = (const float*)d_in[7];
    float*       out      = (float*)d_out;

    const int n_dst   = in_sizes[1] / D;   // 20000
    const int n_edges = in_sizes[2];       // 640000

    float* summed = (float*)d_ws;                       // n_dst * 128 floats
    float* deg    = summed + (size_t)n_dst * D;         // n_dst floats
    const int ws_elems = n_dst * D + n_dst;             // divisible by 4

    // Phase 0: zero accumulators (every call; deterministic).
    zero_ws_kernel<<<2560, 256, 0, stream>>>((float4*)d_ws, ws_elems / 4);

    // Phase 1: gather + atomic segment-sum. 8 edges (waves) per 256-thr block.
    const int scatter_blocks = (n_edges + 7) / 8;
    edge_scatter_kernel<<<scatter_blocks, 256, 0, stream>>>(
        feat_src, src_idx, dst_idx, summed, deg, n_edges);

    // Phase 2: WMMA GEMM + bias. 1250 blocks x 8 waves (one 16x16 tile each).
    const int gemm_blocks = n_dst / 16;
    sage_wmma_kernel<<<gemm_blocks, 256, 0, stream>>>(
        feat_dst, summed, deg, W_self, b_self, W_neigh, b_neigh, out);
}